// K1GPUModel_27307402067995
// MI455X (gfx1250) — compile-verified
//
#include <hip/hip_runtime.h>

#define VOCAB 32000
#define EMBED 128
#define HIDDEN 256
#define MAX_AGENTS 2000
#define MAX_CHILDREN 4
#define NDEPTH 4      // MAX_DEPTH + 1 forward iterations
#define NTOK 2048     // B * T

typedef __attribute__((ext_vector_type(2))) float v2f;
typedef __attribute__((ext_vector_type(8))) float v8f;

// -------------------------------------------------------------------------
// Kernel A: hierarchical routing MLP. One 256-thread block per token.
// Weights are L2-resident (<=85 distinct agents ~ 22MB); GEMV reads are
// fully coalesced (thread j reads column j, stride-1 across the block).
// -------------------------------------------------------------------------
__global__ __launch_bounds__(256) void route_mlp_kernel(
    const int*   __restrict__ xidx,
    const float* __restrict__ emb,
    const float* __restrict__ W1, const float* __restrict__ b1,
    const float* __restrict__ W2, const float* __restrict__ b2,
    const float* __restrict__ Wr, const float* __restrict__ br,
    const int*   __restrict__ children,
    float*       __restrict__ act /* [NTOK, EMBED] */)
{
    __shared__ float xs[EMBED];
    __shared__ float hs[HIDDEN];
    __shared__ float rr[MAX_CHILDREN];
    __shared__ int   curr_s;

    const int tok = blockIdx.x;
    const int tid = threadIdx.x;

    // Embedding gather for this token
    const int v = xidx[tok];
    if (tid < EMBED) xs[tid] = emb[(long)v * EMBED + tid];
    if (tid == 0) curr_s = 0;
    __syncthreads();

    float out_e = 0.f;
    for (int it = 0; it < NDEPTH; ++it) {
        const int a = curr_s;

        // h[tid] = relu( sum_e x[e] * W1[a][e][tid] + b1[a][tid] )
        {
            const float* w1 = W1 + (long)a * EMBED * HIDDEN + tid;
            float acc = b1[(long)a * HIDDEN + tid];
            #pragma unroll 8
            for (int e = 0; e < EMBED; ++e)
                acc = fmaf(xs[e], w1[(long)e * HIDDEN], acc);
            hs[tid] = acc > 0.f ? acc : 0.f;
        }
        __syncthreads();

        // out[tid] = sum_j h[j] * W2[a][j][tid] + b2[a][tid]   (threads 0..127)
        if (tid < EMBED) {
            const float* w2 = W2 + (long)a * HIDDEN * EMBED + tid;
            float oe = b2[(long)a * EMBED + tid];
            #pragma unroll 8
            for (int j = 0; j < HIDDEN; ++j)
                oe = fmaf(hs[j], w2[(long)j * EMBED], oe);
            out_e = oe;
        }
        // router logits (threads 0..3)
        if (tid < MAX_CHILDREN) {
            const float* wr = Wr + (long)a * HIDDEN * MAX_CHILDREN + tid;
            float rl = br[(long)a * MAX_CHILDREN + tid];
            #pragma unroll 8
            for (int j = 0; j < HIDDEN; ++j)
                rl = fmaf(hs[j], wr[(long)j * MAX_CHILDREN], rl);
            rr[tid] = rl;
        }
        __syncthreads();

        // argmax (first-max tie-break like jnp.argmax) + child lookup
        if (tid == 0) {
            int best = 0; float bv = rr[0];
            #pragma unroll
            for (int c = 1; c < MAX_CHILDREN; ++c)
                if (rr[c] > bv) { bv = rr[c]; best = c; }
            int na = children[a * MAX_CHILDREN + best];
            if (na != -1) curr_s = na;
        }
        __syncthreads();
    }
    // only the final iteration's `out` survives (matches reference)
    if (tid < EMBED) act[(long)tok * EMBED + tid] = out_e;
}

// -------------------------------------------------------------------------
// Kernel B: logits = act[2048,128] @ Wo[128,32000] + bo, fp32 WMMA.
// Store-bandwidth bound (262MB out); fp32 16x16x4 WMMA keeps exact numerics.
// One wave per 16x64 output tile: 4 accumulators, A reused across 4 B tiles.
// A layout : lane<16 -> M=lane, K=k{0,1}; lane>=16 -> K=k{2,3}  (v2f load)
// B layout : mirrors A over K; N = n0 + lane%16
// C/D      : VGPR r, lane<16 -> M=r, lane>=16 -> M=8+r
// -------------------------------------------------------------------------
__global__ __launch_bounds__(256) void gemm_logits_wmma(
    const float* __restrict__ act,   // [NTOK, EMBED]
    const float* __restrict__ Wo,    // [EMBED, VOCAB]
    const float* __restrict__ bo,    // [VOCAB]
    float*       __restrict__ logits)// [NTOK, VOCAB]
{
    const int wave   = (int)((blockIdx.x * blockDim.x + threadIdx.x) >> 5);
    const int lane   = threadIdx.x & 31;
    const int NT_M   = NTOK / 16;          // 128 M-tiles
    const int mTile  = wave % NT_M;
    const int nChunk = wave / NT_M;        // 0..499
    const int m0 = mTile * 16;
    const int n0 = nChunk * 64;
    const int lm = lane & 15;
    const int lh = lane >> 4;

    v8f acc0, acc1, acc2, acc3;
    {
        const float bz0 = bo[n0 + lm];
        const float bz1 = bo[n0 + 16 + lm];
        const float bz2 = bo[n0 + 32 + lm];
        const float bz3 = bo[n0 + 48 + lm];
        #pragma unroll
        for (int r = 0; r < 8; ++r) {
            acc0[r] = bz0; acc1[r] = bz1; acc2[r] = bz2; acc3[r] = bz3;
        }
    }

    const float* arow  = act + (long)(m0 + lm) * EMBED + 2 * lh;
    const float* bcol0 = Wo + (long)(2 * lh) * VOCAB + n0 + lm;

    #pragma unroll 4
    for (int k = 0; k < EMBED; k += 4) {
        v2f a = *(const v2f*)(arow + k);              // K = k+2*lh, k+2*lh+1
        const float* bp = bcol0 + (long)k * VOCAB;
        v2f b0, b1v, b2v, b3v;
        b0.x  = bp[0];       b0.y  = bp[VOCAB];
        b1v.x = bp[16];      b1v.y = bp[VOCAB + 16];
        b2v.x = bp[32];      b2v.y = bp[VOCAB + 32];
        b3v.x = bp[48];      b3v.y = bp[VOCAB + 48];
        acc0 = __builtin_amdgcn_wmma_f32_16x16x4_f32(false, a, false, b0,  (short)0, acc0, false, false);
        acc1 = __builtin_amdgcn_wmma_f32_16x16x4_f32(false, a, false, b1v, (short)0, acc1, false, false);
        acc2 = __builtin_amdgcn_wmma_f32_16x16x4_f32(false, a, false, b2v, (short)0, acc2, false, false);
        acc3 = __builtin_amdgcn_wmma_f32_16x16x4_f32(false, a, false, b3v, (short)0, acc3, false, false);
    }

    float* orow = logits + (long)(m0 + 8 * lh) * VOCAB + n0 + lm;
    #pragma unroll
    for (int r = 0; r < 8; ++r) {
        float* op = orow + (long)r * VOCAB;
        op[0]  = acc0[r];
        op[16] = acc1[r];
        op[32] = acc2[r];
        op[48] = acc3[r];
    }
}

extern "C" void kernel_launch(void* const* d_in, const int* in_sizes, int n_in,
                              void* d_out, int out_size, void* d_ws, size_t ws_size,
                              hipStream_t stream) {
    const int*   xidx     = (const int*)d_in[0];
    const float* emb      = (const float*)d_in[1];
    const float* W1       = (const float*)d_in[2];
    const float* b1       = (const float*)d_in[3];
    const float* W2       = (const float*)d_in[4];
    const float* b2       = (const float*)d_in[5];
    const float* Wr       = (const float*)d_in[6];
    const float* br       = (const float*)d_in[7];
    const float* Wo       = (const float*)d_in[8];
    const float* bo       = (const float*)d_in[9];
    const int*   children = (const int*)d_in[10];

    float* logits = (float*)d_out;
    float* act    = (float*)d_ws;   // [NTOK * EMBED] fp32 scratch (1 MB)

    route_mlp_kernel<<<NTOK, 256, 0, stream>>>(
        xidx, emb, W1, b1, W2, b2, Wr, br, children, act);

    // 128 M-tiles * 500 N-chunks = 64000 waves; 8 waves per 256-thread block
    gemm_logits_wmma<<<8000, 256, 0, stream>>>(act, Wo, bo, logits);
}